// MMDLoss_34471407518033
// MI455X (gfx1250) — compile-verified
//
#include <hip/hip_runtime.h>

#define GAMMA_F 0.1f
#define BM 128
#define BN 128
#define BK 32
#define LDK 36  // padded LDS row stride: 16B-aligned rows + conflict-free b64 gathers

typedef float v2f __attribute__((ext_vector_type(2)));
typedef float v8f __attribute__((ext_vector_type(8)));

__global__ void zero_out_kernel(float* out) { out[0] = 0.0f; }

// One wave (32 threads) per row: squared L2 norm of each row of z1 (rows 0..N-1)
// and z2 (rows N..N+M-1) into norms[].
__global__ __launch_bounds__(32)
void row_norm_kernel(const float* __restrict__ z1, const float* __restrict__ z2,
                     float* __restrict__ norms, int N, int M, int D)
{
    int row = blockIdx.x;
    const float* zr = (row < N) ? (z1 + (size_t)row * D)
                                : (z2 + (size_t)(row - N) * D);
    float s = 0.0f;
    for (int c = threadIdx.x; c < D; c += 32) {
        float v = zr[c];
        s += v * v;
    }
#pragma unroll
    for (int off = 16; off > 0; off >>= 1) s += __shfl_xor(s, off, 32);
    if (threadIdx.x == 0) norms[row] = s;
}

// weight * sum_{i,j} exp(-gamma*(nA_i + nB_j - 2*Za_i.Zb_j)) * bA_i * bB_j -> atomicAdd(out).
// 128x128 block tile, 8 waves in 4x2 grid, each wave 32x64 (8 f32 WMMA accumulators).
// symmetric==1: only lower-triangle blocks run; off-diagonal blocks count double.
__global__ __launch_bounds__(256)
void mmd_pair_kernel(const float* __restrict__ Za, const float* __restrict__ Zb,
                     const float* __restrict__ nA, const float* __restrict__ nB,
                     const float* __restrict__ bA, const float* __restrict__ bB,
                     float weight, int symmetric, int nRows, int nCols, int D,
                     float* __restrict__ out)
{
    __shared__ float As[BM * LDK];
    __shared__ float Bs[BN * LDK];

    int bx = blockIdx.x;   // column tile (Zb rows)
    int by = blockIdx.y;   // row tile (Za rows)
    if (symmetric) {
        if (bx > by) return;          // block-uniform -> EXEC all-ones below
        if (bx < by) weight *= 2.0f;  // mirrored block has identical sum
    }
    int rowBase = by * BM;
    int colBase = bx * BN;

    int tid  = threadIdx.x;
    int lane = tid & 31;
    int wid  = tid >> 5;    // 8 waves
    int wm   = wid & 3;     // 4 waves along M: rows wm*32 .. +32
    int wn   = wid >> 2;    // 2 waves along N: cols wn*64 .. +64
    int hi   = lane >> 4;   // lane half (K-pair select for f32 WMMA fragments)
    int lr   = lane & 15;

    v8f acc[2][4];
#pragma unroll
    for (int t = 0; t < 2; ++t)
#pragma unroll
        for (int u = 0; u < 4; ++u)
            acc[t][u] = (v8f){0.f,0.f,0.f,0.f,0.f,0.f,0.f,0.f};

    // Staging: 128x32 tile = 1024 float4; 256 threads -> 4 float4 each (per matrix).
    // idx = tid + j*256 ; r = idx>>3 ; c4 = idx&7 (column group of 4 floats).
    float4 ra[4], rb[4];

    auto load_stage = [&](int k0) {
#pragma unroll
        for (int j = 0; j < 4; ++j) {
            int idx = tid + j * 256;
            int r  = idx >> 3;
            int c4 = idx & 7;
            int kc = k0 + c4 * 4;
            int rra = rowBase + r; rra = (rra < nRows) ? rra : (nRows - 1);
            int rrb = colBase + r; rrb = (rrb < nCols) ? rrb : (nCols - 1);
            const float* pa = Za + (size_t)rra * D + kc;
            const float* pb = Zb + (size_t)rrb * D + kc;
            if (kc + 3 < D) {
                ra[j] = *(const float4*)pa;
                rb[j] = *(const float4*)pb;
            } else {
                ra[j].x = (kc + 0 < D) ? pa[0] : 0.f;
                ra[j].y = (kc + 1 < D) ? pa[1] : 0.f;
                ra[j].z = (kc + 2 < D) ? pa[2] : 0.f;
                ra[j].w = (kc + 3 < D) ? pa[3] : 0.f;
                rb[j].x = (kc + 0 < D) ? pb[0] : 0.f;
                rb[j].y = (kc + 1 < D) ? pb[1] : 0.f;
                rb[j].z = (kc + 2 < D) ? pb[2] : 0.f;
                rb[j].w = (kc + 3 < D) ? pb[3] : 0.f;
            }
        }
    };

    load_stage(0);

    for (int k0 = 0; k0 < D; k0 += BK) {
        // Commit register buffers to LDS (b128 stores, 16B aligned).
#pragma unroll
        for (int j = 0; j < 4; ++j) {
            int idx = tid + j * 256;
            int r  = idx >> 3;
            int c4 = idx & 7;
            *(float4*)&As[r * LDK + c4 * 4] = ra[j];
            *(float4*)&Bs[r * LDK + c4 * 4] = rb[j];
        }
        __syncthreads();

        // Prefetch next stage into registers; overlaps with WMMA loop below.
        if (k0 + BK < D) load_stage(k0 + BK);

        // 8 x K=4 WMMA steps. f32 WMMA A frag: lane l holds row (l&15),
        // K = (l>>4)*2 + {0,1}; B frag mirrors with columns = Zb rows.
#pragma unroll
        for (int ks = 0; ks < BK; ks += 4) {
            int c = ks + hi * 2;
            v2f a[2], b[4];
#pragma unroll
            for (int t = 0; t < 2; ++t) {
                const float* ap = &As[(wm * 32 + t * 16 + lr) * LDK + c];
                a[t].x = ap[0]; a[t].y = ap[1];
            }
#pragma unroll
            for (int u = 0; u < 4; ++u) {
                const float* bp = &Bs[(wn * 64 + u * 16 + lr) * LDK + c];
                b[u].x = bp[0]; b[u].y = bp[1];
            }
#pragma unroll
            for (int t = 0; t < 2; ++t)
#pragma unroll
                for (int u = 0; u < 4; ++u)
                    acc[t][u] = __builtin_amdgcn_wmma_f32_16x16x4_f32(
                        false, a[t], false, b[u], (short)0, acc[t][u], false, false);
        }
        __syncthreads();
    }

    // Fused epilogue. C/D layout: VGPR j, lane l -> M = (l>>4)*8 + j, N = l&15.
    float nb[4], bb[4];
#pragma unroll
    for (int u = 0; u < 4; ++u) {
        int col = colBase + wn * 64 + u * 16 + lr;
        bool ok = (col < nCols);
        nb[u] = ok ? nB[col] : 0.0f;
        bb[u] = ok ? bB[col] : 0.0f;
    }

    float s = 0.0f;
#pragma unroll
    for (int t = 0; t < 2; ++t) {
        int rbase = rowBase + wm * 32 + t * 16 + hi * 8;
#pragma unroll
        for (int j = 0; j < 8; ++j) {
            int row = rbase + j;
            bool rok = (row < nRows);
            float na = rok ? nA[row] : 0.0f;
            float ba = rok ? bA[row] : 0.0f;
            float partial = 0.0f;
#pragma unroll
            for (int u = 0; u < 4; ++u) {
                float e = __expf(GAMMA_F * (2.0f * acc[t][u][j] - na - nb[u]));
                partial += bb[u] * e;
            }
            s += ba * partial;
        }
    }
#pragma unroll
    for (int off = 16; off > 0; off >>= 1) s += __shfl_xor(s, off, 32);
    if (lane == 0) atomicAdd(out, weight * s);
}

extern "C" void kernel_launch(void* const* d_in, const int* in_sizes, int n_in,
                              void* d_out, int out_size, void* d_ws, size_t ws_size,
                              hipStream_t stream)
{
    const float* z1 = (const float*)d_in[0];
    const float* z2 = (const float*)d_in[1];
    const float* b1 = (const float*)d_in[2];
    const float* b2 = (const float*)d_in[3];
    int N = in_sizes[2];           // 4096
    int M = in_sizes[3];           // 4096
    int D = in_sizes[0] / N;       // 512
    float* out = (float*)d_out;

    float* norms = (float*)d_ws;   // N + M floats of scratch
    float* nA = norms;
    float* nB = norms + N;

    zero_out_kernel<<<1, 1, 0, stream>>>(out);
    row_norm_kernel<<<N + M, 32, 0, stream>>>(z1, z2, norms, N, M, D);

    dim3 blk(256);
    dim3 g11((N + BN - 1) / BN, (N + BM - 1) / BM);
    dim3 g12((M + BN - 1) / BN, (N + BM - 1) / BM);
    dim3 g22((M + BN - 1) / BN, (M + BM - 1) / BM);

    // term1 - 2*term2 + term3, all accumulated atomically into out[0]
    mmd_pair_kernel<<<g11, blk, 0, stream>>>(z1, z1, nA, nA, b1, b1,  1.0f, 1, N, N, D, out);
    mmd_pair_kernel<<<g12, blk, 0, stream>>>(z1, z2, nA, nB, b1, b2, -2.0f, 0, N, M, D, out);
    mmd_pair_kernel<<<g22, blk, 0, stream>>>(z2, z2, nB, nB, b2, b2,  1.0f, 1, M, M, D, out);
}